// MLZS_87041807220943
// MI455X (gfx1250) — compile-verified
//
#include <hip/hip_runtime.h>
#include <math.h>

typedef __attribute__((ext_vector_type(2))) float v2f;
typedef __attribute__((ext_vector_type(8))) float v8f;

#define WMMA_F32(a, b, c) \
  __builtin_amdgcn_wmma_f32_16x16x4_f32(false, (a), false, (b), (short)0, (c), false, false)

// Problem constants
#define Bn 64
#define Sn 512
#define En 300
#define Ln 2000
#define NFn 50
#define HQn 256
#define FSn 10
#define SPn 503   // S - FS + 1

enum { F_ACC = 1, F_RELU = 2, F_BIAS_ROW = 4, F_BIAS_COL = 8 };

// ---------------------------------------------------------------------------
// Generic batched fp32 WMMA GEMM: C[M,N] (+)= A[M,K] * B[K,N] (+bias)(relu)
// One wave computes a 16x64 group (4 adjacent 16x16 n-tiles): each A fragment
// is loaded once and feeds 4 V_WMMA_F32_16X16X4_F32 ops.
// BT: B is stored transposed, element B[k,n] = Bm[n*ldb + k].
// Hot-loop loads are unconditional (clamped addresses) + v_cndmask select.
// ---------------------------------------------------------------------------
template <bool BT>
__global__ void gemm_f32_wmma(const float* __restrict__ A, int lda, long sA,
                              const float* __restrict__ Bm, int ldb, long sB,
                              float* __restrict__ C, int ldc, long sC,
                              int M, int N, int K,
                              const float* __restrict__ bias, int flags)
{
  const int wid  = threadIdx.x >> 5;
  const int lane = threadIdx.x & 31;
  const int ntn  = (N + 15) >> 4;
  const int ntm  = (M + 15) >> 4;
  const int gn   = (ntn + 3) >> 2;             // groups of 4 n-tiles
  const int grp  = blockIdx.x * (blockDim.x >> 5) + wid;
  if (grp >= ntm * gn) return;                 // wave-uniform exit
  const int mt  = grp / gn;
  const int ng0 = (grp - mt * gn) * 4;         // first n-tile of this group

  const float* Ab = A  + (long)blockIdx.y * sA;
  const float* Bb = Bm + (long)blockIdx.y * sB;
  float*       Cb = C  + (long)blockIdx.y * sC;

  const int hi2 = (lane >> 4) * 2;             // K-pair selector: 0 or 2
  const int lr  = lane & 15;
  const int am  = mt * 16 + lr;                // A row held by this lane
  const bool amv = am < M;
  const float* Ap = Ab + (long)(amv ? am : 0) * lda + hi2;

  int  bn[4];
  bool bnv[4];
  const float* Bp[4];
  v8f acc[4];
  const v8f z = {0.f, 0.f, 0.f, 0.f, 0.f, 0.f, 0.f, 0.f};
#pragma unroll
  for (int t = 0; t < 4; ++t) {
    bn[t]  = (ng0 + t) * 16 + lr;              // B col held by this lane
    bnv[t] = bn[t] < N;
    const int bnc = bnv[t] ? bn[t] : 0;
    Bp[t] = BT ? (Bb + (long)bnc * ldb + hi2)  // row bnc of B^T
               : (Bb + (long)hi2 * ldb + bnc); // col bnc of B
    acc[t] = z;
  }

  const int Kf = K & ~3;
  int k0 = 0;
#pragma unroll 2
  for (; k0 < Kf; k0 += 4) {
    v2f a;
    float a0 = Ap[k0], a1 = Ap[k0 + 1];
    a.x = amv ? a0 : 0.f;
    a.y = amv ? a1 : 0.f;
#pragma unroll
    for (int t = 0; t < 4; ++t) {
      v2f b;
      float b0, b1;
      if (BT) { b0 = Bp[t][k0];             b1 = Bp[t][k0 + 1]; }
      else    { b0 = Bp[t][(long)k0 * ldb]; b1 = Bp[t][(long)(k0 + 1) * ldb]; }
      b.x = bnv[t] ? b0 : 0.f;
      b.y = bnv[t] ? b1 : 0.f;
      acc[t] = WMMA_F32(a, b, acc[t]);
    }
  }
  if (k0 < K) {                                // tail (K % 4 != 0), one step
    const int ka  = k0 + hi2;
    const bool kv0 = ka < K, kv1 = ka + 1 < K;
    const int t0 = kv0 ? k0 : k0 - 2;          // clamp within valid K range
    const int t1 = kv1 ? k0 + 1 : (kv0 ? k0 : k0 - 2);
    v2f a;
    float a0 = Ap[t0], a1 = Ap[t1];
    a.x = (amv && kv0) ? a0 : 0.f;
    a.y = (amv && kv1) ? a1 : 0.f;
#pragma unroll
    for (int t = 0; t < 4; ++t) {
      v2f b;
      float b0, b1;
      if (BT) { b0 = Bp[t][t0];             b1 = Bp[t][t1]; }
      else    { b0 = Bp[t][(long)t0 * ldb]; b1 = Bp[t][(long)t1 * ldb]; }
      b.x = (bnv[t] && kv0) ? b0 : 0.f;
      b.y = (bnv[t] && kv1) ? b1 : 0.f;
      acc[t] = WMMA_F32(a, b, acc[t]);
    }
  }
#pragma unroll
  for (int t = 0; t < 4; ++t) {
#pragma unroll
    for (int i = 0; i < 8; ++i) {
      const int m = mt * 16 + i + hi2 * 4;     // C/D layout: VGPR i -> M=i / M=8+i
      if (m < M && bnv[t]) {
        float v = acc[t][i];
        if (flags & F_BIAS_ROW) v += bias[m];
        if (flags & F_BIAS_COL) v += bias[bn[t]];
        if (flags & F_ACC)      v += Cb[(long)m * ldc + bn[t]];
        if (flags & F_RELU)     v  = fmaxf(v, 0.f);
        Cb[(long)m * ldc + bn[t]] = v;
      }
    }
  }
}

// ---------------------------------------------------------------------------
// Row softmax for the [L,L] adjacency matrices.
// ---------------------------------------------------------------------------
__global__ void softmax_rows(const float* __restrict__ in, float* __restrict__ out, int ncols)
{
  __shared__ float red[256];
  const long row = blockIdx.x;
  const float* r = in  + row * ncols;
  float*       o = out + row * ncols;
  const int t = threadIdx.x;

  float mx = -INFINITY;
  for (int c = t; c < ncols; c += 256) mx = fmaxf(mx, r[c]);
  red[t] = mx; __syncthreads();
  for (int s = 128; s > 0; s >>= 1) { if (t < s) red[t] = fmaxf(red[t], red[t + s]); __syncthreads(); }
  mx = red[0]; __syncthreads();

  float sum = 0.f;
  for (int c = t; c < ncols; c += 256) sum += __expf(r[c] - mx);
  red[t] = sum; __syncthreads();
  for (int s = 128; s > 0; s >>= 1) { if (t < s) red[t] += red[t + s]; __syncthreads(); }
  const float inv = 1.f / red[0];
  for (int c = t; c < ncols; c += 256) o[c] = __expf(r[c] - mx) * inv;
}

// ---------------------------------------------------------------------------
// Fused attention: per (batch, 16-label tile)
//   strip[16,503] = label_mat[l0:l0+16,:] @ DsqT[b]      (WMMA, K=300)
//   softmax over s (seq dim) per label row                (in LDS)
//   c_att[16,50]  = strip @ D[b][503,50]                  (WMMA, K=504 padded)
// Phase 1 keeps K as the OUTER loop with the 4 n-tiles inner, so each A
// fragment (read from LDS) feeds 4 WMMAs immediately. The 16x300 label_mat
// strip is staged in LDS once per block (8 waves share identical A fragments).
// ---------------------------------------------------------------------------
__global__ void attn_fused(const float* __restrict__ label_mat,
                           const float* __restrict__ DsqT,   // [B, E, S']
                           const float* __restrict__ D,      // [B, S', NF]
                           float* __restrict__ c_att)        // [B, L, NF]
{
  __shared__ float strip[16 * 512];
  __shared__ float aLDS[16 * 304];                          // 16 x 300, padded stride
  const int b   = blockIdx.y;
  const int l0  = blockIdx.x * 16;
  const int tid = threadIdx.x;
  const int wid = tid >> 5, lane = tid & 31;
  const int hi2 = (lane >> 4) * 2, lr = lane & 15;

  // Stage A strip (L=2000=125*16: no row guard needed)
  for (int idx = tid; idx < 16 * En; idx += 256) {
    const int rr = idx / En, cc = idx - rr * En;
    aLDS[rr * 304 + cc] = label_mat[(long)(l0 + rr) * En + cc];
  }
  __syncthreads();

  const float* Bpb = DsqT + (long)b * (En * SPn);

  // Phase 1: scores. Each of 8 waves covers 4 of the 32 n-tiles (512 cols).
  {
    bool nv[4];
    const float* Bp[4];
    v8f acc[4];
    const v8f z = {0.f, 0.f, 0.f, 0.f, 0.f, 0.f, 0.f, 0.f};
#pragma unroll
    for (int t = 0; t < 4; ++t) {
      const int n = (wid + 8 * t) * 16 + lr;
      nv[t]  = n < SPn;
      Bp[t]  = Bpb + (long)hi2 * SPn + (nv[t] ? n : 0);
      acc[t] = z;
    }
    const float* Ar = aLDS + lr * 304 + hi2;
#pragma unroll 2
    for (int k0 = 0; k0 < En; k0 += 4) {                  // E=300 multiple of 4
      v2f a;
      a.x = Ar[k0];
      a.y = Ar[k0 + 1];
#pragma unroll
      for (int t = 0; t < 4; ++t) {
        v2f bb;
        float b0 = Bp[t][(long)k0 * SPn];
        float b1 = Bp[t][(long)(k0 + 1) * SPn];
        bb.x = nv[t] ? b0 : 0.f;
        bb.y = nv[t] ? b1 : 0.f;
        acc[t] = WMMA_F32(a, bb, acc[t]);
      }
    }
#pragma unroll
    for (int t = 0; t < 4; ++t)
#pragma unroll
      for (int i = 0; i < 8; ++i)
        strip[(i + hi2 * 4) * 512 + (wid + 8 * t) * 16 + lr] = acc[t][i];
  }
  __syncthreads();

  // Phase 2: softmax over s in [0,503) for row r; 16-lane group reductions.
  {
    const int r = 2 * wid + (hi2 >> 1);
    float mx = -INFINITY;
    for (int j = 0; j < 32; ++j) { int idx = lr + 16 * j; if (idx < SPn) mx = fmaxf(mx, strip[r * 512 + idx]); }
    for (int off = 8; off > 0; off >>= 1) mx = fmaxf(mx, __shfl_xor(mx, off, 16));
    float s = 0.f;
    for (int j = 0; j < 32; ++j) {
      int idx = lr + 16 * j;
      if (idx < SPn) { float e = __expf(strip[r * 512 + idx] - mx); strip[r * 512 + idx] = e; s += e; }
      else           { strip[r * 512 + idx] = 0.f; }      // zero-pad cols 503..511 for phase 3
    }
    for (int off = 8; off > 0; off >>= 1) s += __shfl_xor(s, off, 16);
    const float inv = 1.f / s;
    for (int j = 0; j < 32; ++j) { int idx = lr + 16 * j; if (idx < SPn) strip[r * 512 + idx] *= inv; }
  }
  __syncthreads();

  // Phase 3: c_att tile. Waves 0..3 cover the 4 n-tiles of NF=50.
  if (wid < 4) {
    const int n  = wid * 16 + lr;
    const bool nv = n < NFn;
    const float* Db = D + (long)b * (SPn * NFn) + (nv ? n : 0);
    v8f acc = {0.f, 0.f, 0.f, 0.f, 0.f, 0.f, 0.f, 0.f};
#pragma unroll 2
    for (int k0 = 0; k0 < 504; k0 += 4) {                 // strip zero-padded to 504
      const int ka = k0 + hi2;
      v2f a, bb;
      a.x = strip[lr * 512 + ka];                         // cols 503..511 are zero
      a.y = strip[lr * 512 + ka + 1];
      const int kc0 = ka     < SPn ? ka     : SPn - 1;    // clamp, select below
      const int kc1 = ka + 1 < SPn ? ka + 1 : SPn - 1;
      float b0 = Db[(long)kc0 * NFn];
      float b1 = Db[(long)kc1 * NFn];
      bb.x = (nv && ka     < SPn) ? b0 : 0.f;
      bb.y = (nv && ka + 1 < SPn) ? b1 : 0.f;
      acc = WMMA_F32(a, bb, acc);
    }
#pragma unroll
    for (int i = 0; i < 8; ++i) {
      const int m = i + hi2 * 4;
      if (nv) c_att[((long)b * Ln + l0 + m) * NFn + n] = acc[i];
    }
  }
}

// ---------------------------------------------------------------------------
// Final fusion: res[b,l] = sum_d lm3[l,d] * relu(c_att[b,l,:]·dm_w[d,:] + dm_b[d])
// One wave per (b,l); lanes split the 556 d's; wave shuffle reduction.
// ---------------------------------------------------------------------------
__global__ void final_fuse(const float* __restrict__ c_att,
                           const float* __restrict__ dm_w, const float* __restrict__ dm_b,
                           const float* __restrict__ label_mat, const float* __restrict__ lm2,
                           float* __restrict__ out)
{
  const int wid = threadIdx.x >> 5, lane = threadIdx.x & 31;
  const long p = (long)blockIdx.x * 8 + wid;              // p = b*L + l
  if (p >= (long)Bn * Ln) return;
  const int l = (int)(p % Ln);
  const float* cr = c_att + p * NFn;
  float accum = 0.f;
  for (int d = lane; d < HQn + En; d += 32) {
    const float* wr = dm_w + (long)d * NFn;
    float dot = dm_b[d];
#pragma unroll 10
    for (int f = 0; f < NFn; ++f) dot += cr[f] * wr[f];
    const float e   = fmaxf(dot, 0.f);
    const float lm3 = (d < En) ? label_mat[(long)l * En + d] : lm2[(long)l * HQn + (d - En)];
    accum += lm3 * e;
  }
  for (int off = 16; off > 0; off >>= 1) accum += __shfl_xor(accum, off, 32);
  if (lane == 0) out[p] = accum;
}

// ---------------------------------------------------------------------------
static inline int gemm_blocks(int M, int N)
{
  const int ntm = (M + 15) / 16;
  const int gn  = ((N + 15) / 16 + 3) / 4;    // groups of 4 n-tiles
  return (ntm * gn + 7) / 8;                  // 8 waves per block
}

extern "C" void kernel_launch(void* const* d_in, const int* in_sizes, int n_in,
                              void* d_out, int out_size, void* d_ws, size_t ws_size,
                              hipStream_t stream)
{
  (void)in_sizes; (void)n_in; (void)out_size; (void)ws_size;
  const float* x       = (const float*)d_in[0];
  const float* label   = (const float*)d_in[1];
  const float* adjP_in = (const float*)d_in[2];
  const float* adjC_in = (const float*)d_in[3];
  const float* conv_w  = (const float*)d_in[4];
  const float* conv_b  = (const float*)d_in[5];
  const float* sq_w    = (const float*)d_in[6];
  const float* sq_b    = (const float*)d_in[7];
  const float* dm_w    = (const float*)d_in[8];
  const float* dm_b    = (const float*)d_in[9];
  const float* g1_ws   = (const float*)d_in[10];
  const float* g1_wp   = (const float*)d_in[11];
  const float* g1_wc   = (const float*)d_in[12];
  const float* g1_b    = (const float*)d_in[13];
  const float* g2_ws   = (const float*)d_in[14];
  const float* g2_wp   = (const float*)d_in[15];
  const float* g2_wc   = (const float*)d_in[16];
  const float* g2_b    = (const float*)d_in[17];
  float* out = (float*)d_out;

  float* w = (float*)d_ws;
  float* adjP = w; w += (long)Ln * Ln;        // 16 MB
  float* adjC = w; w += (long)Ln * Ln;        // 16 MB
  float* Dbuf = w; w += (long)Bn * SPn * NFn; // 6.4 MB  [B,S',NF]
  float* DsqT = w; w += (long)Bn * En * SPn;  // 38.6 MB [B,E,S']
  float* cAtt = w; w += (long)Bn * Ln * NFn;  // 25.6 MB [B,L,NF]
  float* tP   = w; w += (long)Ln * En;
  float* tC   = w; w += (long)Ln * En;
  float* lm1  = w; w += (long)Ln * HQn;
  float* tP1  = w; w += (long)Ln * HQn;
  float* tC1  = w; w += (long)Ln * HQn;
  float* lm2  = w; w += (long)Ln * HQn;

  // adjacency softmaxes
  softmax_rows<<<Ln, 256, 0, stream>>>(adjP_in, adjP, Ln);
  softmax_rows<<<Ln, 256, 0, stream>>>(adjC_in, adjC, Ln);

  // Conv as GEMM: per batch, A = x[b] with lda=E (overlapping contiguous patches),
  // M=503, K=3000, B = conv_w^T (BT, ldb=3000), +conv_b (col bias) -> D [B,S',NF]
  gemm_f32_wmma<true><<<dim3(gemm_blocks(SPn, NFn), Bn), 256, 0, stream>>>(
      x, En, (long)Sn * En, conv_w, FSn * En, 0, Dbuf, NFn, (long)SPn * NFn,
      SPn, NFn, FSn * En, conv_b, F_BIAS_COL);

  // D_square (transposed): DsqT[b] = sq_w[300,50] @ D[b]^T[50,503] + sq_b (row bias)
  gemm_f32_wmma<true><<<dim3(gemm_blocks(En, SPn), Bn), 256, 0, stream>>>(
      sq_w, NFn, 0, Dbuf, NFn, (long)SPn * NFn, DsqT, SPn, (long)En * SPn,
      En, SPn, NFn, sq_b, F_BIAS_ROW);

  // GCN layer 1
  {
    dim3 gA(gemm_blocks(Ln, En), 1);    // [2000,2000]@[2000,300]
    gemm_f32_wmma<false><<<gA, 256, 0, stream>>>(adjP, Ln, 0, label, En, 0, tP, En, 0, Ln, En, Ln, nullptr, 0);
    gemm_f32_wmma<false><<<gA, 256, 0, stream>>>(adjC, Ln, 0, label, En, 0, tC, En, 0, Ln, En, Ln, nullptr, 0);
    dim3 gB(gemm_blocks(Ln, HQn), 1);   // [2000,300]@[300,256]
    gemm_f32_wmma<false><<<gB, 256, 0, stream>>>(label, En, 0, g1_ws, HQn, 0, lm1, HQn, 0, Ln, HQn, En, nullptr, 0);
    gemm_f32_wmma<false><<<gB, 256, 0, stream>>>(tP, En, 0, g1_wp, HQn, 0, lm1, HQn, 0, Ln, HQn, En, nullptr, F_ACC);
    gemm_f32_wmma<false><<<gB, 256, 0, stream>>>(tC, En, 0, g1_wc, HQn, 0, lm1, HQn, 0, Ln, HQn, En, g1_b,
                                                 F_ACC | F_BIAS_COL | F_RELU);
  }
  // GCN layer 2
  {
    dim3 gA(gemm_blocks(Ln, HQn), 1);   // [2000,2000]@[2000,256] and [2000,256]@[256,256]
    gemm_f32_wmma<false><<<gA, 256, 0, stream>>>(adjP, Ln, 0, lm1, HQn, 0, tP1, HQn, 0, Ln, HQn, Ln, nullptr, 0);
    gemm_f32_wmma<false><<<gA, 256, 0, stream>>>(adjC, Ln, 0, lm1, HQn, 0, tC1, HQn, 0, Ln, HQn, Ln, nullptr, 0);
    gemm_f32_wmma<false><<<gA, 256, 0, stream>>>(lm1, HQn, 0, g2_ws, HQn, 0, lm2, HQn, 0, Ln, HQn, HQn, nullptr, 0);
    gemm_f32_wmma<false><<<gA, 256, 0, stream>>>(tP1, HQn, 0, g2_wp, HQn, 0, lm2, HQn, 0, Ln, HQn, HQn, nullptr, F_ACC);
    gemm_f32_wmma<false><<<gA, 256, 0, stream>>>(tC1, HQn, 0, g2_wc, HQn, 0, lm2, HQn, 0, Ln, HQn, HQn, g2_b,
                                                 F_ACC | F_BIAS_COL | F_RELU);
  }

  // fused attention (scores -> seq softmax -> c_att), then final diagonal fusion
  attn_fused<<<dim3(Ln / 16, Bn), 256, 0, stream>>>(label, DsqT, Dbuf, cAtt);
  final_fuse<<<dim3((Bn * Ln) / 8), 256, 0, stream>>>(cAtt, dm_w, dm_b, label, lm2, out);
}